// SpectralConv2D_10617159155840
// MI455X (gfx1250) — compile-verified
//
#include <hip/hip_runtime.h>
#include <cstdint>
#include <cstddef>

// SpectralConv2D (FNO layer), N=16 R=128 IN=64 OUT=64 MODES=16, f32.
// Truncated-spectrum decomposition into 5 WMMA GEMM kernels + 1 weight repack.
// All matrix math on v_wmma_f32_16x16x4_f32; twiddles served from LDS tables.

#define BN   16      // batch
#define RR   128     // spatial
#define IND  64      // in channels
#define OUTD 64      // out channels
#define MD   16      // modes kept per axis (32 h-rows total: 0..15 and 112..127)

typedef __attribute__((ext_vector_type(2))) float v2f;
typedef __attribute__((ext_vector_type(8))) float v8f;

#define ANG128 0.049087385212340526f   /* 2*pi/128 */
#define NORMF  0.0078125f              /* 1/128 (ortho: 1/128 fwd, 1/128 inv) */

__device__ __forceinline__ v8f wmma4(v2f a, v2f b, v8f c) {
  // D = A(16x4 f32) * B(4x16 f32) + C(16x16 f32)
  return __builtin_amdgcn_wmma_f32_16x16x4_f32(false, a, false, b, (short)0, c,
                                               false, false);
}

// ---------------------------------------------------------------------------
// Kernel 0: repack spectral weights for coalesced access in the mix stage.
// src: fw[i][o][hh][w][re/im]   dst: Wr/Wi[(h*16+w)][i][o]   (h: 0..31)
// ---------------------------------------------------------------------------
__global__ __launch_bounds__(256) void k_repack_w(const float* __restrict__ fw0,
                                                  const float* __restrict__ fw1,
                                                  float* __restrict__ Wr,
                                                  float* __restrict__ Wi) {
  const size_t t = (size_t)blockIdx.x * 256 + threadIdx.x;  // 32*16*64*64 entries
  const int o = (int)(t & 63);
  size_t r = t >> 6;
  const int i = (int)(r & 63);  r >>= 6;
  const int w = (int)(r & 15);  r >>= 4;
  const int h = (int)r;                         // 0..31
  const float* fw = (h < MD) ? fw0 : fw1;
  const int hh = (h < MD) ? h : h - MD;
  const size_t src = ((((size_t)i * OUTD + o) * MD + hh) * MD + w) * 2;
  Wr[t] = fw[src];
  Wi[t] = fw[src + 1];
}

// ---------------------------------------------------------------------------
// Kernel 1: x-DFT, w = 0..15.  Per (n,y): S1[w,i] = (1/128) * sum_x e^{-i th} X[x,i]
// GEMM M=16(w) N=64(i) K=128(x).  Blocks: 2048 (n*y), 128 thr = 4 waves (i-tiles).
// S1 layout: [n][w][y][i].  Twiddle tables pre-scaled by 1/128.
// ---------------------------------------------------------------------------
__global__ __launch_bounds__(128) void k_fft_x(const float* __restrict__ X,
                                               float* __restrict__ S1r,
                                               float* __restrict__ S1i) {
  __shared__ float tc[128], tn[128];   // cos/128, -sin/128
  {
    const int t = threadIdx.x;         // blockDim == 128
    float s, c;
    __sincosf((float)t * ANG128, &s, &c);
    tc[t] = c * NORMF;
    tn[t] = -s * NORMF;
  }
  __syncthreads();

  const int ny = blockIdx.x;
  const int n = ny >> 7, y = ny & 127;
  const int wave = threadIdx.x >> 5, lane = threadIdx.x & 31;
  const int hi = lane >> 4, l15 = lane & 15;
  const int i0 = wave * 16;
  const float* Xs = X + ((size_t)(n * RR + y)) * RR * IND;  // [x=128][i=64]

  v8f cr = {}; v8f ci = {};
  for (int k0 = 0; k0 < RR; k0 += 4) {
    const int k = k0 + 2 * hi;  // x index for this lane-half
    const int p0 = (l15 * k) & 127, p1 = (l15 * (k + 1)) & 127;
    v2f ac;  ac.x  = tc[p0]; ac.y  = tc[p1];
    v2f ans; ans.x = tn[p0]; ans.y = tn[p1];
    v2f b;   b.x = Xs[(size_t)k * IND + i0 + l15];
             b.y = Xs[(size_t)(k + 1) * IND + i0 + l15];
    cr = wmma4(ac,  b, cr);
    ci = wmma4(ans, b, ci);
  }
#pragma unroll
  for (int r = 0; r < 8; ++r) {
    const int w = r + 8 * hi;  // D row
    const size_t idx = (((size_t)(n * MD + w)) * RR + y) * IND + i0 + l15;
    S1r[idx] = cr[r];
    S1i[idx] = ci[r];
  }
}

// ---------------------------------------------------------------------------
// Kernel 2: y-DFT, h' in {0..15, 112..127}.  Per (n,w): F[h,i] = sum_y e^{-i phi} S1[y,i]
// GEMM M=32(h) N=64(i) K=128(y), complex x complex.
// Blocks: 256 (n*w), 256 thr = 8 waves: (h-tile 0..1) x (i-tile 0..3).
// F layout: [h(32)][w][n][i]
// ---------------------------------------------------------------------------
__global__ __launch_bounds__(256) void k_fft_y(const float* __restrict__ S1r,
                                               const float* __restrict__ S1i,
                                               float* __restrict__ Fr,
                                               float* __restrict__ Fi) {
  __shared__ float tc[128], ts[128], tn[128];   // cos, sin, -sin
  if (threadIdx.x < 128) {
    const int t = threadIdx.x;
    float s, c;
    __sincosf((float)t * ANG128, &s, &c);
    tc[t] = c; ts[t] = s; tn[t] = -s;
  }
  __syncthreads();

  const int n = blockIdx.x >> 4, w = blockIdx.x & 15;
  const int wave = threadIdx.x >> 5, lane = threadIdx.x & 31;
  const int hi = lane >> 4, l15 = lane & 15;
  const int mt = wave >> 2, i0 = (wave & 3) * 16;
  const int hrow = mt * 16 + l15;
  const int hact = (hrow < MD) ? hrow : hrow + 96;  // 112..127
  const float* Sr = S1r + ((size_t)(n * MD + w)) * RR * IND;  // [y][i]
  const float* Si = S1i + ((size_t)(n * MD + w)) * RR * IND;

  v8f fr = {}; v8f fi = {};
  for (int k0 = 0; k0 < RR; k0 += 4) {
    const int k = k0 + 2 * hi;  // y
    const int p0 = (hact * k) & 127, p1 = (hact * (k + 1)) & 127;
    v2f ac;  ac.x  = tc[p0]; ac.y  = tc[p1];
    v2f as;  as.x  = ts[p0]; as.y  = ts[p1];
    v2f ans; ans.x = tn[p0]; ans.y = tn[p1];
    v2f br;  br.x = Sr[(size_t)k * IND + i0 + l15];
             br.y = Sr[(size_t)(k + 1) * IND + i0 + l15];
    v2f bi;  bi.x = Si[(size_t)k * IND + i0 + l15];
             bi.y = Si[(size_t)(k + 1) * IND + i0 + l15];
    fr = wmma4(ac,  br, fr);  fr = wmma4(as,  bi, fr);
    fi = wmma4(ac,  bi, fi);  fi = wmma4(ans, br, fi);
  }
#pragma unroll
  for (int r = 0; r < 8; ++r) {
    const int h = mt * 16 + r + 8 * hi;
    const size_t idx = (((size_t)(h * MD + w)) * BN + n) * IND + i0 + l15;
    Fr[idx] = fr[r];
    Fi[idx] = fi[r];
  }
}

// ---------------------------------------------------------------------------
// Kernel 3: channel mixing per frequency.  Per (h,w): G[n,o] = sum_i F[n,i] * W[i,o]
// GEMM M=16(n) N=64(o) K=64(i), complex x complex, repacked coalesced weights.
// Blocks: 512 (h*w), 128 thr = 4 waves (o-tiles).  G layout: [n][o][h(32)][w]
// ---------------------------------------------------------------------------
__global__ __launch_bounds__(128) void k_mix(const float* __restrict__ Fr,
                                             const float* __restrict__ Fi,
                                             const float* __restrict__ Wr,
                                             const float* __restrict__ Wi,
                                             float* __restrict__ Gr,
                                             float* __restrict__ Gi) {
  const int h = blockIdx.x >> 4, w = blockIdx.x & 15;
  const int wave = threadIdx.x >> 5, lane = threadIdx.x & 31;
  const int hi = lane >> 4, l15 = lane & 15;
  const int o0 = wave * 16;
  const float* Ar = Fr + ((size_t)(h * MD + w)) * BN * IND;   // [n=16][i=64]
  const float* Ai = Fi + ((size_t)(h * MD + w)) * BN * IND;
  const float* Br = Wr + ((size_t)(h * MD + w)) * IND * OUTD; // [i=64][o=64]
  const float* Bi = Wi + ((size_t)(h * MD + w)) * IND * OUTD;

  v8f gr = {}; v8f gi = {};
  for (int k0 = 0; k0 < IND; k0 += 4) {
    const int k = k0 + 2 * hi;  // i
    v2f far;  far.x = Ar[l15 * IND + k];  far.y = Ar[l15 * IND + k + 1];
    v2f fai;  fai.x = Ai[l15 * IND + k];  fai.y = Ai[l15 * IND + k + 1];
    v2f fain; fain.x = -fai.x;            fain.y = -fai.y;
    v2f wrf; wrf.x = Br[(size_t)k * OUTD + o0 + l15];
             wrf.y = Br[(size_t)(k + 1) * OUTD + o0 + l15];
    v2f wif; wif.x = Bi[(size_t)k * OUTD + o0 + l15];
             wif.y = Bi[(size_t)(k + 1) * OUTD + o0 + l15];
    gr = wmma4(far,  wrf, gr);  gr = wmma4(fain, wif, gr);
    gi = wmma4(far,  wif, gi);  gi = wmma4(fai,  wrf, gi);
  }
#pragma unroll
  for (int r = 0; r < 8; ++r) {
    const int nrow = r + 8 * hi;  // batch index
    const size_t idx = (((size_t)(nrow * OUTD + o0 + l15)) * 32 + h) * MD + w;
    Gr[idx] = gr[r];
    Gi[idx] = gi[r];
  }
}

// ---------------------------------------------------------------------------
// Kernel 4: inverse y.  Per (n,o): T[y,w] = sum_h e^{+i phi(h',y)} G[h,w]
// GEMM M=128(y) N=16(w) K=32(h), complex x complex.
// Blocks: 1024 (n*o), 256 thr = 8 waves (y-tiles).  T layout: [n][y][w][o]
// ---------------------------------------------------------------------------
__global__ __launch_bounds__(256) void k_ifft_y(const float* __restrict__ Gr,
                                                const float* __restrict__ Gi,
                                                float* __restrict__ Tr,
                                                float* __restrict__ Ti) {
  __shared__ float tc[128], ts[128], tn[128];
  if (threadIdx.x < 128) {
    const int t = threadIdx.x;
    float s, c;
    __sincosf((float)t * ANG128, &s, &c);
    tc[t] = c; ts[t] = s; tn[t] = -s;
  }
  __syncthreads();

  const int n = blockIdx.x >> 6, o = blockIdx.x & 63;
  const int wave = threadIdx.x >> 5, lane = threadIdx.x & 31;
  const int hi = lane >> 4, l15 = lane & 15;
  const int m0 = wave * 16;
  const int yrow = m0 + l15;
  const float* Br = Gr + ((size_t)(n * OUTD + o)) * 32 * MD;  // [h=32][w=16]
  const float* Bi = Gi + ((size_t)(n * OUTD + o)) * 32 * MD;

  v8f tr = {}; v8f ti = {};
#pragma unroll
  for (int k0 = 0; k0 < 32; k0 += 4) {
    const int k = k0 + 2 * hi;  // h
    const int h0a = (k < MD) ? k : k + 96;
    const int h1a = (k + 1 < MD) ? k + 1 : k + 97;
    const int p0 = (h0a * yrow) & 127, p1 = (h1a * yrow) & 127;
    v2f ac;  ac.x  = tc[p0]; ac.y  = tc[p1];
    v2f as;  as.x  = ts[p0]; as.y  = ts[p1];
    v2f ans; ans.x = tn[p0]; ans.y = tn[p1];
    v2f br;  br.x = Br[k * MD + l15];  br.y = Br[(k + 1) * MD + l15];
    v2f bi;  bi.x = Bi[k * MD + l15];  bi.y = Bi[(k + 1) * MD + l15];
    tr = wmma4(ac,  br, tr);  tr = wmma4(ans, bi, tr);
    ti = wmma4(ac,  bi, ti);  ti = wmma4(as,  br, ti);
  }
#pragma unroll
  for (int r = 0; r < 8; ++r) {
    const int y = m0 + r + 8 * hi;
    const size_t idx = (((size_t)(n * RR + y)) * MD + l15) * OUTD + o;
    Tr[idx] = tr[r];
    Ti[idx] = ti[r];
  }
}

// ---------------------------------------------------------------------------
// Kernel 5 (fused): inverse x (hermitian fold) + residual GEMM + bias + SiLU
// Per (n,y): Y[x,o] = sum_w c(w)/128 * (cos*Tr - sin*Ti)  +  X[x,:]@w_res  + b_res
// GEMM M=128(x) N=64(o) K=16(w) + K=64(i).  Blocks: 2048 (n*y), 256 thr = 8 waves.
// Tables pre-scaled by 2/128 (fold factor); w=0 column fixed up with x0.5.
// ---------------------------------------------------------------------------
__global__ __launch_bounds__(256) void k_ifft_x_fused(const float* __restrict__ X,
                                                      const float* __restrict__ w_res,
                                                      const float* __restrict__ b_res,
                                                      const float* __restrict__ Tr,
                                                      const float* __restrict__ Ti,
                                                      float* __restrict__ out) {
  __shared__ float tc[128], tn[128];   // 2*cos/128, -2*sin/128
  if (threadIdx.x < 128) {
    const int t = threadIdx.x;
    float s, c;
    __sincosf((float)t * ANG128, &s, &c);
    tc[t] = c * (2.0f * NORMF);
    tn[t] = -s * (2.0f * NORMF);
  }
  __syncthreads();

  const int ny = blockIdx.x;
  const int n = ny >> 7, y = ny & 127;
  const int wave = threadIdx.x >> 5, lane = threadIdx.x & 31;
  const int hi = lane >> 4, l15 = lane & 15;
  const int m0 = wave * 16;
  const int xrow = m0 + l15;
  const float* Xs  = X  + ((size_t)(n * RR + y)) * RR * IND;   // [x][i]
  const float* Tsr = Tr + ((size_t)(n * RR + y)) * MD * OUTD;  // [w=16][o=64]
  const float* Tsi = Ti + ((size_t)(n * RR + y)) * MD * OUTD;

  for (int nt = 0; nt < 4; ++nt) {
    const int o0 = nt * 16;
    v8f acc = {};
    // spectral part: K = 16 (w)
#pragma unroll
    for (int k0 = 0; k0 < MD; k0 += 4) {
      const int k = k0 + 2 * hi;  // w
      const float fix = (k == 0) ? 0.5f : 1.0f;  // c(0)=1, c(w>0)=2
      const int p0 = (k * xrow) & 127, p1 = ((k + 1) * xrow) & 127;
      v2f ac;  ac.x  = tc[p0] * fix; ac.y  = tc[p1];
      v2f ans; ans.x = tn[p0] * fix; ans.y = tn[p1];
      v2f br;  br.x = Tsr[k * OUTD + o0 + l15];
               br.y = Tsr[(k + 1) * OUTD + o0 + l15];
      v2f bi;  bi.x = Tsi[k * OUTD + o0 + l15];
               bi.y = Tsi[(k + 1) * OUTD + o0 + l15];
      acc = wmma4(ac,  br, acc);
      acc = wmma4(ans, bi, acc);
    }
    // residual GEMM: K = 64 (i)
    for (int k0 = 0; k0 < IND; k0 += 4) {
      const int k = k0 + 2 * hi;  // i
      v2f a; a.x = Xs[(size_t)xrow * IND + k];
             a.y = Xs[(size_t)xrow * IND + k + 1];
      v2f b; b.x = w_res[(size_t)k * OUTD + o0 + l15];
             b.y = w_res[(size_t)(k + 1) * OUTD + o0 + l15];
      acc = wmma4(a, b, acc);
    }
    // epilogue: bias + SiLU
#pragma unroll
    for (int r = 0; r < 8; ++r) {
      const int x = m0 + r + 8 * hi;
      const int o = o0 + l15;
      const float v = acc[r] + b_res[o];
      const float s = v / (1.0f + __expf(-v));
      out[(((size_t)(n * RR + y)) * RR + x) * OUTD + o] = s;
    }
  }
}

// ---------------------------------------------------------------------------
extern "C" void kernel_launch(void* const* d_in, const int* in_sizes, int n_in,
                              void* d_out, int out_size, void* d_ws, size_t ws_size,
                              hipStream_t stream) {
  const float* X     = (const float*)d_in[0];  // [16,128,128,64]
  const float* w_res = (const float*)d_in[1];  // [64,64]
  const float* b_res = (const float*)d_in[2];  // [64]
  const float* fw0   = (const float*)d_in[3];  // [64,64,16,16,2]
  const float* fw1   = (const float*)d_in[4];  // [64,64,16,16,2]
  float* out = (float*)d_out;                  // [16,128,128,64]

  // workspace partition (floats); total 14,680,064 floats = 56 MiB
  float* ws  = (float*)d_ws;
  float* S1r = ws;                       // 16*16*128*64 = 2,097,152
  float* S1i = S1r + 2097152;
  float* Fr  = S1i + 2097152;            // 32*16*16*64  =   524,288
  float* Fi  = Fr  + 524288;
  float* Gr  = Fi  + 524288;             // 16*64*32*16  =   524,288
  float* Gi  = Gr  + 524288;
  float* Tr  = Gi  + 524288;             // 16*128*16*64 = 2,097,152
  float* Ti  = Tr  + 2097152;
  float* Wr  = Ti  + 2097152;            // 32*16*64*64  = 2,097,152
  float* Wi  = Wr  + 2097152;

  k_repack_w    <<<8192,     256, 0, stream>>>(fw0, fw1, Wr, Wi);
  k_fft_x       <<<BN * RR,  128, 0, stream>>>(X, S1r, S1i);
  k_fft_y       <<<BN * MD,  256, 0, stream>>>(S1r, S1i, Fr, Fi);
  k_mix         <<<32 * MD,  128, 0, stream>>>(Fr, Fi, Wr, Wi, Gr, Gi);
  k_ifft_y      <<<BN * OUTD, 256, 0, stream>>>(Gr, Gi, Tr, Ti);
  k_ifft_x_fused<<<BN * RR,  256, 0, stream>>>(X, w_res, b_res, Tr, Ti, out);
}